// HungarianLoss_69870527971728
// MI455X (gfx1250) — compile-verified
//
#include <hip/hip_runtime.h>
#include <hip/hip_bf16.h>

// Problem constants (match reference).
static constexpr int kB = 128;   // batches
static constexpr int kN = 128;   // nodes (square assignment)
static constexpr int kD = 512;   // feature dim
static constexpr float kINF = 1e9f;

typedef __attribute__((ext_vector_type(2))) float v2f;
typedef __attribute__((ext_vector_type(8))) float v8f;

// ---------------------------------------------------------------------------
// Kernel 0: zero the scalar output (harness poisons d_out with 0xAA).
// ---------------------------------------------------------------------------
__global__ void zero_out_kernel(float* out) {
    if (threadIdx.x == 0 && blockIdx.x == 0) out[0] = 0.0f;
}

// ---------------------------------------------------------------------------
// Kernel 1: per-row sums of squares for pred (x2) and target (y2).
// One thread per row, float4 loads (128 b128 loads per thread).
// ---------------------------------------------------------------------------
__global__ __launch_bounds__(256) void sumsq_kernel(const float* __restrict__ X,
                                                    const float* __restrict__ Y,
                                                    float* __restrict__ x2,
                                                    float* __restrict__ y2) {
    int idx = blockIdx.x * 256 + threadIdx.x;      // row index in [0, B*N)
    if (idx >= kB * kN) return;
    const float4* px = (const float4*)(X + (size_t)idx * kD);
    const float4* py = (const float4*)(Y + (size_t)idx * kD);
    float sx = 0.0f, sy = 0.0f;
    #pragma unroll 4
    for (int k = 0; k < kD / 4; ++k) {
        float4 a = px[k];
        float4 c = py[k];
        sx += a.x * a.x + a.y * a.y + a.z * a.z + a.w * a.w;
        sy += c.x * c.x + c.y * c.y + c.z * c.z + c.w * c.w;
    }
    x2[idx] = sx;
    y2[idx] = sy;
}

// ---------------------------------------------------------------------------
// Kernel 2: batched cost GEMM via fp32 WMMA (V_WMMA_F32_16X16X4_F32).
// Grid: (B, N/16, N/16); one wave32 per 16x16 output tile.
// cost[b][m][n] = sqrt(max(s^2*(x2[m] + y2[n] - 2*dot(X[m],Y[n])), 0))
// ---------------------------------------------------------------------------
__global__ __launch_bounds__(32) void cost_wmma_kernel(const float* __restrict__ X,
                                                       const float* __restrict__ Y,
                                                       const float* __restrict__ x2,
                                                       const float* __restrict__ y2,
                                                       const float* __restrict__ fw,
                                                       float* __restrict__ cost) {
    const int b    = blockIdx.x;
    const int tm   = blockIdx.y;         // 0..7, M tile
    const int tn   = blockIdx.z;         // 0..7, N tile
    const int lane = threadIdx.x;        // 0..31 (wave32)
    const int half = lane >> 4;          // 0 or 1
    const int l16  = lane & 15;

    const float s  = fw[0];
    const float s2 = s * s;

    // A fragment (16x4 f32): lane -> row M = l16, K = k0 + half*2 + {0,1}
    // B fragment (4x16 f32): lane -> col N = l16, K = k0 + half*2 + {0,1}
    const float* arow = X + ((size_t)b * kN + tm * 16 + l16) * kD + half * 2;
    const float* brow = Y + ((size_t)b * kN + tn * 16 + l16) * kD + half * 2;

    v8f acc = {};
    #pragma unroll 8
    for (int k0 = 0; k0 < kD; k0 += 4) {
        v2f a  = *(const v2f*)(arow + k0);
        v2f bb = *(const v2f*)(brow + k0);
        // (neg_a, A, neg_b, B, c_mod, C, reuse_a, reuse_b)
        acc = __builtin_amdgcn_wmma_f32_16x16x4_f32(
            false, a, false, bb, (short)0, acc, false, false);
    }

    // Epilogue: C/D layout -> VGPR r holds M = r + half*8, N = l16.
    const float ysq = s2 * y2[(size_t)b * kN + tn * 16 + l16];
    float* ctile = cost + ((size_t)b * kN + tm * 16) * kN + tn * 16;
    #pragma unroll
    for (int r = 0; r < 8; ++r) {
        int m = r + half * 8;
        float xsq = s2 * x2[(size_t)b * kN + tm * 16 + m];
        float d2  = xsq + ysq - 2.0f * s2 * acc[r];
        ctile[(size_t)m * kN + l16] = __builtin_sqrtf(fmaxf(d2, 0.0f));
    }
}

// ---------------------------------------------------------------------------
// Kernel 3: Jonker-Volgenant Hungarian, one wave32 per batch.
// Lane-strided ownership of the 129 columns; shuffle-butterfly argmin with
// lowest-index tie-break (matches jnp.argmin).
// ---------------------------------------------------------------------------
__global__ __launch_bounds__(32) void hungarian_kernel(const float* __restrict__ cost,
                                                       int* __restrict__ colout) {
    const int b    = blockIdx.x;
    const int lane = threadIdx.x;
    const float* C = cost + (size_t)b * kN * kN;

    __shared__ float u[kN + 1];
    __shared__ float v[kN + 1];
    __shared__ float minv[kN + 1];
    __shared__ int   p[kN + 1];
    __shared__ int   way[kN + 1];
    __shared__ int   used[kN + 1];

    for (int j = lane; j <= kN; j += 32) {
        u[j] = 0.0f; v[j] = 0.0f; p[j] = 0; way[j] = 0;
    }
    __syncthreads();

    for (int i = 1; i <= kN; ++i) {
        if (lane == 0) p[0] = i;
        for (int j = lane; j <= kN; j += 32) {
            minv[j] = kINF;
            used[j] = 0;
        }
        __syncthreads();

        int j0 = 0;
        while (p[j0] != 0) {
            if (lane == 0) used[j0] = 1;
            const int   i0  = p[j0];
            const float ui0 = u[i0];
            __syncthreads();

            // Update reduced costs for owned columns; track local masked min.
            float best  = kINF;
            int   bestj = kN + 1;
            for (int j = lane; j <= kN; j += 32) {
                if (!used[j]) {
                    // j >= 1 here (column 0 is marked used on first pass).
                    float cur = C[(size_t)(i0 - 1) * kN + (j - 1)] - ui0 - v[j];
                    if (cur < minv[j]) { minv[j] = cur; way[j] = j0; }
                    float m = minv[j];
                    if (m < best || (m == best && j < bestj)) { best = m; bestj = j; }
                }
            }
            // Wave32 butterfly argmin, lowest-index tie-break.
            #pragma unroll
            for (int off = 16; off > 0; off >>= 1) {
                float ob = __shfl_xor(best, off, 32);
                int   oj = __shfl_xor(bestj, off, 32);
                if (ob < best || (ob == best && oj < bestj)) { best = ob; bestj = oj; }
            }
            const float delta = best;
            __syncthreads();   // everyone done reading u/v/minv of this round

            // Dual updates. p[j] distinct across used columns -> no write races.
            for (int j = lane; j <= kN; j += 32) {
                if (used[j]) {
                    u[p[j]] = u[p[j]] + delta;
                    v[j]    = v[j] - delta;
                } else {
                    minv[j] = minv[j] - delta;
                }
            }
            __syncthreads();
            j0 = bestj;
        }

        // Augmenting-path reconstruction (serial; lane 0).
        if (lane == 0) {
            int j = j0;
            while (j != 0) {
                int jn = way[j];
                p[j] = p[jn];
                j = jn;
            }
        }
        __syncthreads();
    }

    // col_of_row: column j-1 matched to row p[j]-1.
    for (int j = 1 + lane; j <= kN; j += 32) {
        colout[(size_t)b * kN + (p[j] - 1)] = j - 1;
    }
}

// ---------------------------------------------------------------------------
// Kernel 4: gather matched targets, MSE per batch, atomic sum into d_out.
// One 256-thread block per batch.
// ---------------------------------------------------------------------------
__global__ __launch_bounds__(256) void loss_kernel(const float* __restrict__ pred,
                                                   const float* __restrict__ targ,
                                                   const int* __restrict__ col,
                                                   float* __restrict__ out) {
    const int b = blockIdx.x;
    const int t = threadIdx.x;
    float s = 0.0f;
    for (int i = 0; i < kN; ++i) {
        const int ci = col[(size_t)b * kN + i];
        const float* pr = pred + ((size_t)b * kN + i)  * kD;
        const float* tg = targ + ((size_t)b * kN + ci) * kD;
        for (int d = t; d < kD; d += 256) {
            float diff = pr[d] - tg[d];
            s += diff * diff;
        }
    }
    __shared__ float red[256];
    red[t] = s;
    __syncthreads();
    #pragma unroll
    for (int off = 128; off > 0; off >>= 1) {
        if (t < off) red[t] += red[t + off];
        __syncthreads();
    }
    if (t == 0) atomicAdd(out, red[0] / (float)(kN * kD));
}

// ---------------------------------------------------------------------------
// Launch: zero -> row norms -> WMMA cost -> Hungarian -> loss (stream-ordered)
// ---------------------------------------------------------------------------
extern "C" void kernel_launch(void* const* d_in, const int* in_sizes, int n_in,
                              void* d_out, int out_size, void* d_ws, size_t ws_size,
                              hipStream_t stream) {
    const float* pred = (const float*)d_in[0];   // [B,N,D] f32
    const float* targ = (const float*)d_in[1];   // [B,N,D] f32
    const float* fw   = (const float*)d_in[2];   // [1,1]   f32
    float* out = (float*)d_out;                  // scalar f32

    // Workspace layout: cost [B*N*N] f32 (8 MB) | x2 [B*N] | y2 [B*N] | col [B*N] i32
    char* ws   = (char*)d_ws;
    float* cost = (float*)ws;
    float* x2   = (float*)(ws + (size_t)kB * kN * kN * sizeof(float));
    float* y2   = x2 + (size_t)kB * kN;
    int*   col  = (int*)(y2 + (size_t)kB * kN);

    zero_out_kernel<<<1, 32, 0, stream>>>(out);

    sumsq_kernel<<<(kB * kN + 255) / 256, 256, 0, stream>>>(pred, targ, x2, y2);

    dim3 gcost(kB, kN / 16, kN / 16);            // 128 x 8 x 8 tiles, 1 wave each
    cost_wmma_kernel<<<gcost, 32, 0, stream>>>(pred, targ, x2, y2, fw, cost);

    hungarian_kernel<<<kB, 32, 0, stream>>>(cost, col);

    loss_kernel<<<kB, 256, 0, stream>>>(pred, targ, col, out);
}